// GCTblock_enc_33724083208342
// MI455X (gfx1250) — compile-verified
//
#include <hip/hip_runtime.h>
#include <math.h>

// ---------------------------------------------------------------------------
// GCT block encoder for MI455X (gfx1250), wave32, f32 WMMA (16x16x4).
// Pair-interleaved operand layouts so every WMMA B-fragment is one b64 load.
// ---------------------------------------------------------------------------

typedef float v2f __attribute__((ext_vector_type(2)));
typedef float v8f __attribute__((ext_vector_type(8)));

// D = A(16x4) * B(4x16) + C, all f32.
#define WMMA_F32X4(a, b, c) \
  __builtin_amdgcn_wmma_f32_16x16x4_f32(false, (a), false, (b), (short)0, (c), false, false)

namespace {
constexpr int Bb = 8, Tin = 96, TT = 192, Nn = 128, DIN = 2, DM = 64;
constexpr int NK = 4, TOPK = 3;
constexpr int NB = Nn * Bb;          // 1024 temporal rows
constexpr int NF = TT / 2 + 1;       // 97 rfft bins
constexpr int MAXLEN = 384;          // >= ceil(192/p)*p for any p in [2,192]
constexpr int CUR_ELEMS  = Bb * TT * Nn * DM;

// conv tiling
constexpr int CONV_POS = 32;         // positions per WG (two 16-row M tiles)
constexpr int CH_TAPS = 7;           // taps per K-chunk (7 chunks of K=448)
constexpr int CH_PAIRS = CH_TAPS * 32;  // 224 k-pairs per chunk
constexpr int RS = 96;               // LDS words per k-pair row (96%64==32 -> bank split)
constexpr int BUFW = CH_PAIRS * RS;  // 21504 floats per chunk buffer
constexpr int RSH = 160;             // agcn Hs stride per node-pair (160%64==32)

// workspace layout (float offsets)
constexpr int WS_SSQ  = 0;                         // 2*128*128
constexpr int WS_V    = WS_SSQ + 2 * 128 * 128;    // 5*64*64 (pair-interleaved)
constexpr int WS_W1   = WS_V + 5 * 64 * 64;        // 49*64*64 (pair-interleaved)
constexpr int WS_B1   = WS_W1 + 49 * 64 * 64;      // 64
constexpr int WS_W2   = WS_B1 + 64;                // 49*64*64
constexpr int WS_B2   = WS_W2 + 49 * 64 * 64;      // 64
constexpr int WS_XT   = WS_B2 + 64;                // [NB][192][64]
constexpr int WS_ACC  = WS_XT + NB * TT * DM;      // [NB][192][64]
constexpr int WS_H    = WS_ACC + NB * TT * DM;     // [NB][384][64]
constexpr int WS_AMP  = WS_H + NB * MAXLEN * DM;   // [NB][97]
constexpr int WS_PW   = WS_AMP + NB * NF;          // [NB][3]
constexpr int WS_GAMP = WS_PW + NB * TOPK;         // 128
constexpr int WS_PLAN = WS_GAMP + 128;             // 16 ints
}  // namespace

// ---------------------------------------------------------------------------
// K1: embedding + time projection.
// ---------------------------------------------------------------------------
__global__ void k_embed_pred(const float* __restrict__ x, const float* __restrict__ embW,
                             const float* __restrict__ embB, const float* __restrict__ predW,
                             const float* __restrict__ predB, float* __restrict__ cur,
                             float* __restrict__ xres) {
  int gid = blockIdx.x * blockDim.x + threadIdx.x;
  if (gid >= Bb * TT * Nn) return;
  int n = gid % Nn, bs = gid / Nn, s = bs % TT, b = bs / TT;
  float z0 = 0.f, z1 = 0.f, rs = 0.f;
  const float* pwr = predW + s * Tin;
  for (int t = 0; t < Tin; ++t) {
    float w = pwr[t];
    const float* xp = x + ((size_t)(b * Tin + t) * Nn + n) * DIN;
    z0 += w * xp[0]; z1 += w * xp[1]; rs += w;
  }
  float pb = predB[s];
  float* xr = xres + ((size_t)(b * TT + s) * Nn + n) * DIN;
  xr[0] = z0 + pb; xr[1] = z1 + pb;
  float* cp = cur + ((size_t)(b * TT + s) * Nn + n) * DM;
  for (int c = 0; c < DM; ++c)
    cp[c] = z0 * embW[c] + z1 * embW[DM + c] + embB[c] * rs + pb;
}

// ---------------------------------------------------------------------------
// K2: S@S for both supports.
// ---------------------------------------------------------------------------
__global__ void k_ssq(const float* __restrict__ S, float* __restrict__ ssq) {
  __shared__ float row[Nn];
  int mat = blockIdx.x >> 7, r = blockIdx.x & 127;
  const float* Sm = S + mat * Nn * Nn;
  row[threadIdx.x] = Sm[r * Nn + threadIdx.x];
  __syncthreads();
  int c = threadIdx.x;
  float a = 0.f;
  for (int m = 0; m < Nn; ++m) a += row[m] * Sm[m * Nn + c];
  ssq[(mat * Nn + r) * Nn + c] = a;
}

// ---------------------------------------------------------------------------
// K3: per-layer weight prep -> pair-interleaved layouts.
//  V[j]: element (krow, col) -> V[j*4096 + (krow>>1)*128 + col*2 + (krow&1)]
//  w{1,2}m: K = tau*64+cin; (K, cout) -> w[(K>>1)*128 + cout*2 + (K&1)]
// ---------------------------------------------------------------------------
__global__ void k_wprep(const float* __restrict__ agcnW, const float* __restrict__ c1W,
                        const float* __restrict__ c1b, const float* __restrict__ c2W,
                        const float* __restrict__ c2b, float* __restrict__ V,
                        float* __restrict__ w1, float* __restrict__ b1,
                        float* __restrict__ w2, float* __restrict__ b2, int layer) {
  int idx = blockIdx.x * blockDim.x + threadIdx.x;
  const int NV = 5 * 64 * 64, NW = 49 * 64 * 64;
  if (idx < NV) {
    int j = idx >> 12, rc = idx & 4095, kr = rc >> 6, c = rc & 63;
    const float* W = agcnW + layer * 384 * 64;
    float v;
    switch (j) {
      case 0: v = W[kr * 64 + c] + W[(192 + kr) * 64 + c]
                  - W[(128 + kr) * 64 + c] - W[(320 + kr) * 64 + c]; break;
      case 1: v = W[(64 + kr) * 64 + c]; break;
      case 2: v = 2.f * W[(128 + kr) * 64 + c]; break;
      case 3: v = W[(256 + kr) * 64 + c]; break;
      default: v = 2.f * W[(320 + kr) * 64 + c]; break;
    }
    V[j * 4096 + (kr >> 1) * 128 + c * 2 + (kr & 1)] = v;
    return;
  }
  idx -= NV;
  if (idx < NW) {
    int o = idx & 63, rest = idx >> 6, i = rest & 63, tau = rest >> 6;
    float s = 0.f;
    for (int k = 0; k < NK; ++k)
      s += c1W[(((size_t)(layer * NK + k) * 64 + o) * 64 + i) * 49 + tau];
    int K = tau * 64 + i;
    w1[(K >> 1) * 128 + o * 2 + (K & 1)] = s * 0.25f;
    return;
  }
  idx -= NW;
  if (idx < 64) {
    float s = 0.f;
    for (int k = 0; k < NK; ++k) s += c1b[(layer * NK + k) * 64 + idx];
    b1[idx] = s * 0.25f; return;
  }
  idx -= 64;
  if (idx < NW) {
    int o = idx & 63, rest = idx >> 6, i = rest & 63, tau = rest >> 6;
    float s = 0.f;
    for (int k = 0; k < NK; ++k)
      s += c2W[(((size_t)(layer * NK + k) * 64 + o) * 64 + i) * 49 + tau];
    int K = tau * 64 + i;
    w2[(K >> 1) * 128 + o * 2 + (K & 1)] = s * 0.25f;
    return;
  }
  idx -= NW;
  if (idx < 64) {
    float s = 0.f;
    for (int k = 0; k < NK; ++k) s += c2b[(layer * NK + k) * 64 + idx];
    b2[idx] = s * 0.25f;
  }
}

// ---------------------------------------------------------------------------
// K4: AGCN, one WG (8 waves) per (b,s) tile X[128,64].
//  H_j (pair-interleaved, stride RSH) staged in 200KB dynamic LDS.
// ---------------------------------------------------------------------------
__global__ __launch_bounds__(256) void k_agcn(const float* __restrict__ cur,
                                              const float* __restrict__ supports,
                                              const float* __restrict__ ssq,
                                              const float* __restrict__ V,
                                              const float* __restrict__ agcnB,
                                              float* __restrict__ xt, int layer) {
  extern __shared__ float Hs[];  // 5 * 64 * RSH floats
  int bs = blockIdx.x;
  int b = bs % Bb, s = bs / Bb;
  const float* X = cur + ((size_t)(b * TT + s)) * Nn * DM;
  int tid = threadIdx.x, wave = tid >> 5, lane = tid & 31;
  int ln = lane & 15, hi = lane >> 4;

  // Phase 1: H_j = X @ V_j, 40 (j, mtile) jobs over 8 waves.
  for (int job = wave; job < 40; job += 8) {
    int j = job >> 3, mt = job & 7;
    v8f acc[4] = {};
    for (int k0 = 0; k0 < DM; k0 += 4) {
      int kb = k0 + hi * 2;
      v2f a = *(const v2f*)(X + (mt * 16 + ln) * DM + kb);
      const float* vb = V + j * 4096 + (kb >> 1) * 128 + ln * 2;
      for (int nt = 0; nt < 4; ++nt) {
        v2f bb = *(const v2f*)(vb + nt * 32);
        acc[nt] = WMMA_F32X4(a, bb, acc[nt]);
      }
    }
    for (int nt = 0; nt < 4; ++nt) {
      int col = nt * 16 + ln;
      for (int v = 0; v < 8; ++v) {
        int node = mt * 16 + v + 8 * hi;
        Hs[j * (64 * RSH) + (node >> 1) * RSH + col * 2 + (node & 1)] = acc[nt][v];
      }
    }
  }
  __syncthreads();

  // Phase 2: out = H_0 + S0 H_1 + S0^2 H_2 + S1 H_3 + S1^2 H_4 + bias.
  int mt = wave;
  v8f acc[4];
  for (int nt = 0; nt < 4; ++nt) {
    int col = nt * 16 + ln;
    for (int v = 0; v < 8; ++v) {
      int row = mt * 16 + v + 8 * hi;
      acc[nt][v] = Hs[(row >> 1) * RSH + col * 2 + (row & 1)];
    }
  }
  const float* Gs[4] = {supports, ssq, supports + Nn * Nn, ssq + Nn * Nn};
  for (int j = 1; j <= 4; ++j) {
    const float* G = Gs[j - 1];
    const float* H = Hs + j * (64 * RSH);
    for (int k0 = 0; k0 < Nn; k0 += 4) {
      int kb = k0 + hi * 2;
      v2f a = *(const v2f*)(G + (mt * 16 + ln) * Nn + kb);
      const float* hb = H + (kb >> 1) * RSH + ln * 2;
      for (int nt = 0; nt < 4; ++nt) {
        v2f bb = *(const v2f*)(hb + nt * 32);
        acc[nt] = WMMA_F32X4(a, bb, acc[nt]);
      }
    }
  }
  for (int nt = 0; nt < 4; ++nt) {
    int col = nt * 16 + ln;
    float bias = agcnB[layer * DM + col];
    for (int v = 0; v < 8; ++v) {
      int node = mt * 16 + v + 8 * hi;
      xt[((size_t)(node * Bb + b) * TT + s) * DM + col] = acc[nt][v] + bias;
    }
  }
}

// ---------------------------------------------------------------------------
// K5: per-row DFT amplitudes; tile pulled into LDS with async-to-LDS copies.
// ---------------------------------------------------------------------------
__global__ __launch_bounds__(256) void k_dft(const float* __restrict__ xt,
                                             float* __restrict__ amp) {
  __shared__ __align__(16) float xs[TT * DM];  // 48KB
  __shared__ float partial[NF * 4];
  int nb = blockIdx.x, tid = threadIdx.x;
  const float* gsrc = xt + (size_t)nb * TT * DM;
  for (int i = tid * 4; i < TT * DM; i += 256 * 4) {
    unsigned ldsa = (unsigned)(size_t)&xs[i];
    unsigned long long ga = (unsigned long long)(size_t)(gsrc + i);
    asm volatile("global_load_async_to_lds_b128 %0, %1, off"
                 :: "v"(ldsa), "v"(ga) : "memory");
  }
  asm volatile("s_wait_asynccnt 0x0" ::: "memory");
  __syncthreads();
  for (int job = tid; job < NF * 4; job += 256) {
    int f = job >> 2, cg = job & 3;
    float re[16] = {}, im[16] = {};
    for (int t = 0; t < TT; ++t) {
      int ft = (f * t) % TT;
      float ang = (float)ft * 0.032724923474893676f;  // pi/96
      float sv, cv;
      sincosf(ang, &sv, &cv);
      const float* xr = xs + t * DM + cg * 16;
      for (int u = 0; u < 16; ++u) { re[u] += xr[u] * cv; im[u] += xr[u] * sv; }
    }
    float sum = 0.f;
    for (int u = 0; u < 16; ++u) sum += sqrtf(re[u] * re[u] + im[u] * im[u]);
    partial[job] = sum;
  }
  __syncthreads();
  for (int f = tid; f < NF; f += 256) {
    float a = (partial[f * 4] + partial[f * 4 + 1] + partial[f * 4 + 2] + partial[f * 4 + 3])
              * (1.f / DM);
    amp[nb * NF + f] = a;
  }
}

// K6: deterministic reduction of amp over rows.
__global__ void k_gamp(const float* __restrict__ amp, float* __restrict__ gamp) {
  __shared__ float red[256];
  int f = blockIdx.x, tid = threadIdx.x;
  float s = 0.f;
  for (int nb = tid; nb < NB; nb += 256) s += amp[nb * NF + f];
  red[tid] = s;
  __syncthreads();
  for (int st = 128; st > 0; st >>= 1) {
    if (tid < st) red[tid] += red[tid + st];
    __syncthreads();
  }
  if (tid == 0) gamp[f] = red[0];
}

// K7: top-3 frequencies (skip DC) -> plan[].
__global__ void k_topk(const float* __restrict__ gamp, int* __restrict__ plan) {
  if (threadIdx.x != 0 || blockIdx.x != 0) return;
  bool used[NF] = {};
  for (int k = 0; k < TOPK; ++k) {
    float best = -1.f; int bi = 1;
    for (int f = 1; f < NF; ++f)
      if (!used[f] && gamp[f] > best) { best = gamp[f]; bi = f; }
    used[bi] = true;
    int p = TT / bi;
    int len = ((TT + p - 1) / p) * p;
    plan[k] = bi; plan[3 + k] = p; plan[6 + k] = len;
  }
}

// K8: per-row softmax over the 3 selected amplitudes.
__global__ void k_pw(const float* __restrict__ amp, const int* __restrict__ plan,
                     float* __restrict__ pw) {
  int nb = blockIdx.x * blockDim.x + threadIdx.x;
  if (nb >= NB) return;
  float a0 = amp[nb * NF + plan[0]];
  float a1 = amp[nb * NF + plan[1]];
  float a2 = amp[nb * NF + plan[2]];
  float m = fmaxf(a0, fmaxf(a1, a2));
  float e0 = expf(a0 - m), e1 = expf(a1 - m), e2 = expf(a2 - m);
  float inv = 1.f / (e0 + e1 + e2);
  pw[nb * 3 + 0] = e0 * inv; pw[nb * 3 + 1] = e1 * inv; pw[nb * 3 + 2] = e2 * inv;
}

// ---------------------------------------------------------------------------
// K9: implicit-GEMM 7x7 SAME conv (64->64 ch), 32 positions per WG.
//  K=3136 staged in 7 double-buffered LDS chunks (2 x 84KB); each weight
//  fragment (one b64) feeds two WMMAs (two position tiles).
//  mode 0: xt -> gelu -> h.   mode 1: h -> acc += pw * (conv + bias).
// ---------------------------------------------------------------------------
__global__ __launch_bounds__(128) void k_conv(const float* __restrict__ xin,
                                              const float* __restrict__ wmat,
                                              const float* __restrict__ bias,
                                              float* __restrict__ outp,
                                              const int* __restrict__ plan,
                                              const float* __restrict__ pw,
                                              int kp, int mode) {
  extern __shared__ float Asm2[];  // 2 * BUFW floats
  int p = plan[3 + kp], L = plan[6 + kp];
  int Hh = L / p;
  int t0 = blockIdx.x * CONV_POS;
  if (t0 >= L) return;  // uniform exit
  int nb = blockIdx.y;
  int inRows = (mode == 0) ? TT : MAXLEN;
  int srcLim = (mode == 0) ? TT : L;
  const float* inb = xin + (size_t)nb * inRows * DM;
  int tid = threadIdx.x;

  auto fill = [&](int ch) {
    float* buf = Asm2 + (ch & 1) * BUFW;
    for (int jj = tid; jj < CH_TAPS * CONV_POS; jj += 128) {
      int tloc = jj >> 5, m = jj & 31;
      int tau = ch * CH_TAPS + tloc;
      int pos = t0 + m;
      int di = tau / 7 - 3, dj = tau % 7 - 3;
      int i = pos / p, j = pos - i * p;
      int si = i + di, sj = j + dj;
      int src = si * p + sj;
      bool ok = (pos < L) && (si >= 0) && (si < Hh) && (sj >= 0) && (sj < p) && (src < srcLim);
      const float* rowp = inb + (size_t)(ok ? src : 0) * DM;
      float* dst = buf + (tloc * 32) * RS + m * 2;
      if (ok) for (int c = 0; c < 64; ++c) dst[(c >> 1) * RS + (c & 1)] = rowp[c];
      else    for (int c = 0; c < 64; ++c) dst[(c >> 1) * RS + (c & 1)] = 0.f;
    }
  };

  int wave = tid >> 5, lane = tid & 31, ln = lane & 15, hi = lane >> 4;
  int col = wave * 16 + ln;  // wave owns one 16-cout tile
  v8f acc0 = {}, acc1 = {};

  fill(0);
  __syncthreads();
  for (int ch = 0; ch < 7; ++ch) {
    if (ch + 1 < 7) {
      fill(ch + 1);
      __builtin_prefetch(wmat + ((size_t)(ch + 1) * CH_PAIRS + lane) * 128, 0, 0);
    }
    const float* bufc = Asm2 + (ch & 1) * BUFW;
    const float* wch = wmat + (size_t)ch * CH_PAIRS * 128 + col * 2;
    for (int kt = 0; kt < CH_TAPS * 16; ++kt) {
      int pair = kt * 2 + hi;
      v2f a0 = *(const v2f*)(bufc + pair * RS + ln * 2);
      v2f a1 = *(const v2f*)(bufc + pair * RS + 32 + ln * 2);
      v2f bb = *(const v2f*)(wch + pair * 128);
      acc0 = WMMA_F32X4(a0, bb, acc0);
      acc1 = WMMA_F32X4(a1, bb, acc1);
    }
    __syncthreads();
  }

  float bv = bias[col];
  for (int half = 0; half < 2; ++half) {
    v8f* A = half ? &acc1 : &acc0;
    int pbase = t0 + half * 16 + 8 * hi;
    if (mode == 0) {
      for (int v = 0; v < 8; ++v) {
        int pos = pbase + v;
        if (pos < L) {
          float xg = (*A)[v] + bv;
          float g = 0.5f * xg * (1.f + erff(xg * 0.70710678118654752f));
          outp[((size_t)nb * MAXLEN + pos) * DM + col] = g;
        }
      }
    } else {
      float wv = pw[nb * 3 + kp];
      for (int v = 0; v < 8; ++v) {
        int pos = pbase + v;
        if (pos < TT) {
          size_t oi = ((size_t)nb * TT + pos) * DM + col;
          outp[oi] += wv * ((*A)[v] + bv);
        }
      }
    }
  }
}

// ---------------------------------------------------------------------------
// K10: LayerNorm over C=64 per (nb,t) -> cur[b,t,n,c].
// ---------------------------------------------------------------------------
__global__ __launch_bounds__(256) void k_ln(const float* __restrict__ accb,
                                            const float* __restrict__ g,
                                            const float* __restrict__ bta,
                                            float* __restrict__ cur) {
  int wave = threadIdx.x >> 5, lane = threadIdx.x & 31;
  int row = blockIdx.x * 8 + wave;  // row = nb*TT + t
  int nb = row / TT, t = row % TT;
  const float* xr = accb + (size_t)row * DM;
  float x0 = xr[lane], x1 = xr[lane + 32];
  float s = x0 + x1;
  for (int m = 16; m; m >>= 1) s += __shfl_xor(s, m, 32);
  float mean = s * (1.f / 64.f);
  float d0 = x0 - mean, d1 = x1 - mean;
  float vs = d0 * d0 + d1 * d1;
  for (int m = 16; m; m >>= 1) vs += __shfl_xor(vs, m, 32);
  float inv = rsqrtf(vs * (1.f / 64.f) + 1e-5f);
  int n = nb / Bb, b = nb % Bb;
  float* op = cur + (((size_t)(b * TT + t)) * Nn + n) * DM;
  op[lane]      = d0 * inv * g[lane] + bta[lane];
  op[lane + 32] = d1 * inv * g[lane + 32] + bta[lane + 32];
}

// ---------------------------------------------------------------------------
extern "C" void kernel_launch(void* const* d_in, const int* in_sizes, int n_in,
                              void* d_out, int out_size, void* d_ws, size_t ws_size,
                              hipStream_t stream) {
  (void)in_sizes; (void)n_in; (void)out_size; (void)ws_size;
  const float* x        = (const float*)d_in[0];
  const float* supports = (const float*)d_in[1];
  const float* embW     = (const float*)d_in[2];
  const float* embB     = (const float*)d_in[3];
  const float* predW    = (const float*)d_in[4];
  const float* predB    = (const float*)d_in[5];
  const float* agcnW    = (const float*)d_in[6];
  const float* agcnB    = (const float*)d_in[7];
  const float* lnG      = (const float*)d_in[8];
  const float* lnB      = (const float*)d_in[9];
  const float* c1W      = (const float*)d_in[10];
  const float* c1b      = (const float*)d_in[11];
  const float* c2W      = (const float*)d_in[12];
  const float* c2b      = (const float*)d_in[13];

  float* cur  = (float*)d_out;
  float* xres = cur + CUR_ELEMS;
  float* ws   = (float*)d_ws;
  float* ssq  = ws + WS_SSQ;
  float* V    = ws + WS_V;
  float* w1m  = ws + WS_W1;
  float* b1m  = ws + WS_B1;
  float* w2m  = ws + WS_W2;
  float* b2m  = ws + WS_B2;
  float* xt   = ws + WS_XT;
  float* accb = ws + WS_ACC;
  float* hbuf = ws + WS_H;
  float* amp  = ws + WS_AMP;
  float* pw   = ws + WS_PW;
  float* gamp = ws + WS_GAMP;
  int*   plan = (int*)(ws + WS_PLAN);

  k_embed_pred<<<(Bb * TT * Nn + 255) / 256, 256, 0, stream>>>(x, embW, embB, predW, predB,
                                                               cur, xres);
  k_ssq<<<256, 128, 0, stream>>>(supports, ssq);

  for (int layer = 0; layer < 2; ++layer) {
    k_wprep<<<(5 * 64 * 64 + 2 * (49 * 64 * 64 + 64) + 255) / 256, 256, 0, stream>>>(
        agcnW, c1W, c1b, c2W, c2b, V, w1m, b1m, w2m, b2m, layer);
    k_agcn<<<TT * Bb, 256, 5 * 64 * RSH * sizeof(float), stream>>>(cur, supports, ssq, V,
                                                                   agcnB, xt, layer);
    k_dft<<<NB, 256, 0, stream>>>(xt, amp);
    k_gamp<<<NF, 256, 0, stream>>>(amp, gamp);
    k_topk<<<1, 1, 0, stream>>>(gamp, plan);
    k_pw<<<(NB + 255) / 256, 256, 0, stream>>>(amp, plan, pw);
    hipMemcpyAsync(accb, xt, sizeof(float) * NB * TT * DM, hipMemcpyDeviceToDevice, stream);
    for (int kp = 0; kp < TOPK; ++kp) {
      k_conv<<<dim3(MAXLEN / CONV_POS, NB), 128, 2 * BUFW * sizeof(float), stream>>>(
          xt, w1m, b1m, hbuf, plan, pw, kp, 0);
      k_conv<<<dim3(MAXLEN / CONV_POS, NB), 128, 2 * BUFW * sizeof(float), stream>>>(
          hbuf, w2m, b2m, accb, plan, pw, kp, 1);
    }
    k_ln<<<NB * TT / 8, 256, 0, stream>>>(accb, lnG, lnB, cur);
  }
}